// ClusteringLoss_55817394979139
// MI455X (gfx1250) — compile-verified
//
#include <hip/hip_runtime.h>

typedef __attribute__((ext_vector_type(2))) float v2f;
typedef __attribute__((ext_vector_type(8))) float v8f;
typedef int v4i_ __attribute__((vector_size(16)));   // matches builtin param pointee

#define N_PTS   8192
#define DIM     512
#define K_CENT  64
#define BM      64            // rows per block (4 waves x 16 rows)
#define WAVES   4
#define DK      128           // D-chunk staged in LDS per iteration
#define LDS_STRIDE 132        // DK + 4 floats pad -> conflict-free b64 reads, 16B-aligned rows
#define EPSF    1e-6f

// ---------------------------------------------------------------------------
// Async global->LDS B128 copy (CDNA5 GLOBAL_LOAD_ASYNC_TO_LDS_B128, ASYNCcnt)
// ---------------------------------------------------------------------------
__device__ __forceinline__ void async_copy_b128(const float* gsrc, float4* lds_dst) {
#if __has_builtin(__builtin_amdgcn_global_load_async_to_lds_b128)
    __builtin_amdgcn_global_load_async_to_lds_b128(
        (__attribute__((address_space(1))) v4i_*)
            (__attribute__((address_space(1))) void*)(void*)const_cast<float*>(gsrc),
        (__attribute__((address_space(3))) v4i_*)
            (__attribute__((address_space(3))) void*)(void*)lds_dst,
        /*imm offset=*/0, /*cpol=*/0);
#else
    // VGLOBAL async encoding: VDST = LDS byte address, VADDR = 64-bit global addr (GV mode)
    unsigned lds_off = (unsigned)(size_t)(__attribute__((address_space(3))) char*)(void*)lds_dst;
    asm volatile("global_load_async_to_lds_b128 %0, %1, off"
                 :: "v"(lds_off), "v"(gsrc)
                 : "memory");
#endif
}

__device__ __forceinline__ void wait_async0() {
#if __has_builtin(__builtin_amdgcn_s_wait_asynccnt)
    __builtin_amdgcn_s_wait_asynccnt(0);
#else
    asm volatile("s_wait_asynccnt 0x0" ::: "memory");
#endif
}

// ---------------------------------------------------------------------------
// Kernel 1: per-centroid sum and sum-of-squares
// ---------------------------------------------------------------------------
__global__ void centroid_stats_kernel(const float* __restrict__ feat,
                                      const int*   __restrict__ cid,
                                      float* __restrict__ sc,
                                      float* __restrict__ sqc) {
    int j = threadIdx.x;
    if (j < K_CENT) {
        const float* row = feat + (size_t)cid[j] * DIM;
        float s = 0.f, q = 0.f;
        for (int d = 0; d < DIM; ++d) { float v = row[d]; s += v; q += v * v; }
        sc[j]  = s;
        sqc[j] = q;
    }
}

// ---------------------------------------------------------------------------
// Kernel 2: N x 64 distance matrix via V_WMMA_F32_16X16X4_F32, fused stats,
//           row-wise min/argmin, per-block energy partial. Centroid tile is
//           gathered into LDS with async B128 copies.
// ---------------------------------------------------------------------------
__global__ void __launch_bounds__(WAVES * 32)
clustering_main_kernel(const float* __restrict__ feat,
                       const int*   __restrict__ cid,
                       const float* __restrict__ sc,
                       const float* __restrict__ sqc,
                       float* __restrict__ y_out,      // length N
                       float* __restrict__ partial) {  // length gridDim.x
    __shared__ float4 ldsC4[K_CENT * (LDS_STRIDE / 4)];   // 64 x 132 floats = 33 KB
    __shared__ float  statsS[WAVES][16];
    __shared__ float  statsQ[WAVES][16];
    __shared__ float  red[WAVES * 2];
    const float* ldsC = (const float*)ldsC4;

    const int tid  = threadIdx.x;
    const int lane = tid & 31;
    const int wv   = tid >> 5;
    const int l    = lane & 15;     // row-in-strip / column-in-tile selector
    const int h    = lane >> 4;     // half-wave selector (K pair 0/1 vs 2/3)
    const int i0   = blockIdx.x * BM + wv * 16;

    const v8f vzero = {0.f, 0.f, 0.f, 0.f, 0.f, 0.f, 0.f, 0.f};
    v8f acc[4] = {vzero, vzero, vzero, vzero};

    float as = 0.f, aq = 0.f;     // per-lane partial row sum / sumsq
    const float* arow = feat + (size_t)(i0 + l) * DIM + 2 * h;

    for (int dk0 = 0; dk0 < DIM; dk0 += DK) {
        __syncthreads();  // previous chunk's readers are done with the LDS tile
        // Gather centroid tile rows [0..63], cols [dk0 .. dk0+DK) into LDS
        // with per-lane async B128 copies (scattered source rows by cid).
        for (int q4 = tid; q4 < K_CENT * (DK / 4); q4 += (int)blockDim.x) {
            int j  = q4 >> 5;       // DK/4 == 32 float4 per row
            int c4 = q4 & 31;
            async_copy_b128(feat + (size_t)cid[j] * DIM + dk0 + c4 * 4,
                            &ldsC4[j * (LDS_STRIDE / 4) + c4]);
        }
        wait_async0();              // this wave's async writes have landed in LDS
        __syncthreads();            // ... and are visible to all waves

        if (dk0 + DK < DIM)         // prefetch next A chunk (global_prefetch_b8)
            __builtin_prefetch(arow + dk0 + DK, 0, 0);

        #pragma unroll 4
        for (int kk = 0; kk < DK; kk += 4) {
            // A fragment (16x4 fp32): lane<16 -> K={kk,kk+1}; lane>=16 -> K={kk+2,kk+3}
            v2f a = *(const v2f*)(arow + dk0 + kk);
            as += a.x + a.y;
            aq += a.x * a.x + a.y * a.y;
            #pragma unroll
            for (int t = 0; t < 4; ++t) {
                // B fragment (4x16): B[k][n] = C[n][k]; same per-lane addressing as A
                v2f b = *(const v2f*)(ldsC + (t * 16 + l) * LDS_STRIDE + kk + 2 * h);
                acc[t] = __builtin_amdgcn_wmma_f32_16x16x4_f32(
                    /*neg_a=*/false, a, /*neg_b=*/false, b,
                    /*c_mod=*/(short)0, acc[t],
                    /*reuse_a=*/false, /*reuse_b=*/false);
            }
        }
    }

    // Combine half-wave stat partials: lane L and L+16 both cover row (L&15).
    as += __shfl_xor(as, 16, 32);
    aq += __shfl_xor(aq, 16, 32);
    if (h == 0) { statsS[wv][l] = as; statsQ[wv][l] = aq; }
    __builtin_amdgcn_wave_barrier();   // same-wave LDS is in-order; keep scheduler honest

    // Preload centroid stats for the 4 columns this lane owns.
    float scv[4], sqcv[4];
    #pragma unroll
    for (int t = 0; t < 4; ++t) { scv[t] = sc[l + 16 * t]; sqcv[t] = sqc[l + 16 * t]; }

    const float deps2 = (float)DIM * EPSF * EPSF;
    float esum = 0.f;

    #pragma unroll
    for (int e = 0; e < 8; ++e) {
        // C/D layout: VGPR e, lanes 0-15 -> M=e ; lanes 16-31 -> M=e+8 ; N = l + 16*t
        const int   m  = e + 8 * h;
        const float sf = statsS[wv][m];
        const float qf = statsQ[wv][m];

        float best  = 3.4e38f;
        int   bestj = 0;
        #pragma unroll
        for (int t = 0; t < 4; ++t) {
            float g    = acc[t][e];
            float d2   = qf + sqcv[t] - 2.f * g + 2.f * EPSF * (sf - scv[t]) + deps2;
            float dist = sqrtf(fmaxf(d2, 0.f));
            int   j    = l + 16 * t;              // ascending -> strict '<' keeps first min
            if (dist < best) { best = dist; bestj = j; }
        }
        // Reduce across the 16 lanes of this half-wave; tie-break to lowest index
        // (matches jnp.argmin first-occurrence semantics), deterministic order.
        #pragma unroll
        for (int mm = 1; mm < 16; mm <<= 1) {
            float ov = __shfl_xor(best,  mm, 32);
            int   oj = __shfl_xor(bestj, mm, 32);
            if (ov < best || (ov == best && oj < bestj)) { best = ov; bestj = oj; }
        }
        if (l == 0) y_out[i0 + m] = (float)bestj;
        esum += best;                              // identical across the half-wave
    }

    if (l == 0) red[wv * 2 + h] = esum;
    __syncthreads();
    if (tid == 0) {
        float t = 0.f;
        for (int i = 0; i < WAVES * 2; ++i) t += red[i];   // fixed order -> deterministic
        partial[blockIdx.x] = t;
    }
}

// ---------------------------------------------------------------------------
// Kernel 3: deterministic final reduction + centroid-row override (last wins,
//           matching JAX scatter-set with duplicate indices).
// ---------------------------------------------------------------------------
__global__ void finalize_kernel(const float* __restrict__ partial, int nblocks,
                                const int* __restrict__ cid,
                                float* __restrict__ out) {
    if (threadIdx.x == 0 && blockIdx.x == 0) {
        float t = 0.f;
        for (int b = 0; b < nblocks; ++b) t += partial[b];
        out[0] = -t;                                   // facility_energy
        for (int k = 0; k < K_CENT; ++k)
            out[1 + cid[k]] = (float)k;                // y_fixed centroid rows
    }
}

// ---------------------------------------------------------------------------
extern "C" void kernel_launch(void* const* d_in, const int* in_sizes, int n_in,
                              void* d_out, int out_size, void* d_ws, size_t ws_size,
                              hipStream_t stream) {
    const float* feat = (const float*)d_in[0];   // (8192, 512) fp32
    const int*   cid  = (const int*)d_in[1];     // (64,) int32
    float* out = (float*)d_out;                  // [0]=energy, [1..N]=y_fixed

    float* wsf     = (float*)d_ws;
    float* sc      = wsf;                        // 64
    float* sqc     = wsf + 64;                   // 64
    float* partial = wsf + 128;                  // N/BM = 128 entries

    const int nblocks = N_PTS / BM;              // 128

    centroid_stats_kernel<<<1, 64, 0, stream>>>(feat, cid, sc, sqc);
    clustering_main_kernel<<<nblocks, WAVES * 32, 0, stream>>>(
        feat, cid, sc, sqc, out + 1, partial);
    finalize_kernel<<<1, 64, 0, stream>>>(partial, nblocks, cid, out);
}